// HetGDC_45853070852383
// MI455X (gfx1250) — compile-verified
//
#include <hip/hip_runtime.h>
#include <hip/hip_bf16.h>

typedef __attribute__((ext_vector_type(2))) float v2f;
typedef __attribute__((ext_vector_type(8))) float v8f;

#define N_NODES   50000
#define D_IN      256
#define D_OUT     128
#define NUM_E     600000
#define NUM_TYPES 4
#define K_HOPS    3
#define ALPHA     0.1f
#define TEMP      0.5f

static_assert(N_NODES % 16 == 0, "M tiling");
static_assert(D_OUT % 16 == 0, "N tiling");
static_assert(D_IN % 4 == 0, "K tiling");

// ---------------------------------------------------------------------------
// 1) H = X * W^T + b  via V_WMMA_F32_16X16X4_F32 (fp32, exact vs reference)
//    grid.x = N/16 row strips; 8 waves per block, wave w owns col tile w*16.
//    Lane layout (ISA 7.12.2, 32-bit A 16x4 / B 4x16):
//      lane L: m_or_n = L%16, K pair = {2*(L/16), 2*(L/16)+1}  -> float2 loads
// ---------------------------------------------------------------------------
__global__ __launch_bounds__(256) void gemm_wmma_f32(
    const float* __restrict__ X, const float* __restrict__ W,
    const float* __restrict__ bias, float* __restrict__ H) {
  const int wave = threadIdx.x >> 5;     // 0..7  -> n-tile
  const int lane = threadIdx.x & 31;
  const int half = lane >> 4;            // K-half selector
  const int r    = lane & 15;            // M for A, N for B
  const int m0   = blockIdx.x * 16;
  const int n0   = wave * 16;

  const float* arow = X + (size_t)(m0 + r) * D_IN + 2 * half;   // A[m][k]
  const float* brow = W + (size_t)(n0 + r) * D_IN + 2 * half;   // B[k][n] = W[n][k]

  v8f acc = {};
#pragma unroll 8
  for (int k0 = 0; k0 < D_IN; k0 += 4) {
    v2f a = *(const v2f*)(arow + k0);
    v2f b = *(const v2f*)(brow + k0);
    acc = __builtin_amdgcn_wmma_f32_16x16x4_f32(
        /*neg_a=*/false, a, /*neg_b=*/false, b,
        /*c_mod=*/(short)0, acc, /*reuse_a=*/false, /*reuse_b=*/false);
  }

  const float bv = bias[n0 + r];
#pragma unroll
  for (int v = 0; v < 8; ++v) {
    // C layout: VGPR v -> row (v + 8*half), col = r
    H[(size_t)(m0 + v + 8 * half) * D_OUT + n0 + r] = acc[v] + bv;
  }
}

// ---------------------------------------------------------------------------
// wave32 reduction helper
// ---------------------------------------------------------------------------
__device__ __forceinline__ float wave_reduce_sum(float v) {
#pragma unroll
  for (int off = 16; off > 0; off >>= 1) v += __shfl_xor(v, off, 32);
  return v;
}

// ---------------------------------------------------------------------------
// 2) row-wise L2 normalize, one wave per row (32 lanes x float4 = 128)
// ---------------------------------------------------------------------------
__global__ __launch_bounds__(256) void l2norm_rows(float* __restrict__ H) {
  const int row  = blockIdx.x * 8 + (threadIdx.x >> 5);
  const int lane = threadIdx.x & 31;
  if (row >= N_NODES) return;
  const size_t off = (size_t)row * D_OUT + 4 * lane;
  float4 h = *(const float4*)(H + off);
  float ss = h.x * h.x + h.y * h.y + h.z * h.z + h.w * h.w;
  ss = wave_reduce_sum(ss);
  const float s = 1.0f / fmaxf(sqrtf(ss), 1e-12f);
  h.x *= s; h.y *= s; h.z *= s; h.w *= s;
  *(float4*)(H + off) = h;
}

// ---------------------------------------------------------------------------
// 3) per-type column sums: block d owns column d (deterministic, no atomics)
// ---------------------------------------------------------------------------
__global__ __launch_bounds__(256) void type_stats(
    const float* __restrict__ H, const int* __restrict__ ntype,
    float* __restrict__ s1, float* __restrict__ s2) {
  const int d = blockIdx.x;
  float a1[NUM_TYPES] = {}, a2[NUM_TYPES] = {};
  for (int i = threadIdx.x; i < N_NODES; i += 256) {
    const int t = ntype[i];
    const float v = H[(size_t)i * D_OUT + d];
    const float vv = v * v;
#pragma unroll
    for (int q = 0; q < NUM_TYPES; ++q) {
      const float m = (t == q) ? 1.0f : 0.0f;
      a1[q] += m * v;
      a2[q] += m * vv;
    }
  }
  __shared__ float l1[NUM_TYPES][256];
  __shared__ float l2s[NUM_TYPES][256];
#pragma unroll
  for (int q = 0; q < NUM_TYPES; ++q) {
    l1[q][threadIdx.x] = a1[q];
    l2s[q][threadIdx.x] = a2[q];
  }
  __syncthreads();
  for (int stride = 128; stride > 0; stride >>= 1) {
    if (threadIdx.x < stride) {
#pragma unroll
      for (int q = 0; q < NUM_TYPES; ++q) {
        l1[q][threadIdx.x]  += l1[q][threadIdx.x + stride];
        l2s[q][threadIdx.x] += l2s[q][threadIdx.x + stride];
      }
    }
    __syncthreads();
  }
  if (threadIdx.x < NUM_TYPES) {
    s1[threadIdx.x * D_OUT + d] = l1[threadIdx.x][0];
    s2[threadIdx.x * D_OUT + d] = l2s[threadIdx.x][0];
  }
}

// 4) mean/std (unbiased). node_type = arange(N) % 4 -> 12500 per type.
__global__ void finalize_stats(const float* __restrict__ s1,
                               const float* __restrict__ s2,
                               float* __restrict__ mean, float* __restrict__ stdv) {
  const int idx = threadIdx.x;  // 0..511 == NUM_TYPES*D_OUT
  const float cnt = (float)(N_NODES / NUM_TYPES);
  const float m = s1[idx] / cnt;
  const float var = (s2[idx] - cnt * m * m) / (cnt - 1.0f);
  mean[idx] = m;
  stdv[idx] = sqrtf(fmaxf(var, 0.0f));
}

// ---------------------------------------------------------------------------
// 5) tilde_H = (H - mean_t)/std_t (in place) + inv row norm for the loss
// ---------------------------------------------------------------------------
__global__ __launch_bounds__(256) void standardize(
    float* __restrict__ H, const int* __restrict__ ntype,
    const float* __restrict__ mean, const float* __restrict__ stdv,
    float* __restrict__ inv_norm) {
  const int row  = blockIdx.x * 8 + (threadIdx.x >> 5);
  const int lane = threadIdx.x & 31;
  if (row >= N_NODES) return;
  const int t = ntype[row];
  const size_t off = (size_t)row * D_OUT + 4 * lane;
  const float4 h = *(const float4*)(H + off);
  const float4 m = *(const float4*)(mean + t * D_OUT + 4 * lane);
  const float4 s = *(const float4*)(stdv + t * D_OUT + 4 * lane);
  float4 td;
  td.x = (h.x - m.x) / s.x;
  td.y = (h.y - m.y) / s.y;
  td.z = (h.z - m.z) / s.z;
  td.w = (h.w - m.w) / s.w;
  *(float4*)(H + off) = td;
  float ss = td.x * td.x + td.y * td.y + td.z * td.z + td.w * td.w;
  ss = wave_reduce_sum(ss);
  if (lane == 0) inv_norm[row] = 1.0f / fmaxf(sqrtf(ss), 1e-12f);
}

// ---------------------------------------------------------------------------
// 6) degree (self loop => deg starts at 1), then dis = rsqrt(deg)
// ---------------------------------------------------------------------------
__global__ void init_misc(float* __restrict__ deg, float* __restrict__ acc) {
  const int i = blockIdx.x * 256 + threadIdx.x;
  if (i < N_NODES) deg[i] = 1.0f;   // self loop contribution
  if (i < 2) acc[i] = 0.0f;         // pos / neg accumulators
}

__global__ void count_deg(const int* __restrict__ col, float* __restrict__ deg) {
  const int e = blockIdx.x * 256 + threadIdx.x;
  if (e < NUM_E) atomicAdd(&deg[col[e]], 1.0f);
}

__global__ void deg_to_dis(float* __restrict__ deg) {
  const int i = blockIdx.x * 256 + threadIdx.x;
  if (i < N_NODES) deg[i] = rsqrtf(deg[i]);  // deg >= 1 always
}

// ---------------------------------------------------------------------------
// 7) APPNP hop:   h_next = alpha*tilde + (1-a)*dis[i]^2*h  (self)  then
//                 h_next[col] += (1-a)*dis[row]*dis[col]*h[row]    (edges)
// ---------------------------------------------------------------------------
__global__ __launch_bounds__(256) void appnp_self(
    const float* __restrict__ tilde, const float* __restrict__ hcur,
    const float* __restrict__ dis, float* __restrict__ hnext) {
  const int idx = blockIdx.x * 256 + threadIdx.x;
  if (idx >= N_NODES * D_OUT) return;
  const int i = idx >> 7;
  const float d = dis[i];
  hnext[idx] = ALPHA * tilde[idx] + (1.0f - ALPHA) * d * d * hcur[idx];
}

__global__ __launch_bounds__(256) void appnp_scatter(
    const int* __restrict__ row, const int* __restrict__ col,
    const float* __restrict__ dis, const float* __restrict__ hcur,
    float* __restrict__ hnext) {
  const int e = blockIdx.x * 8 + (threadIdx.x >> 5);  // one wave per edge
  if (e >= NUM_E) return;
  const int lane = threadIdx.x & 31;
  const int r = row[e], c = col[e];
  const float w = (1.0f - ALPHA) * dis[r] * dis[c];
  const float4 h = *(const float4*)(hcur + (size_t)r * D_OUT + 4 * lane);
  float* dst = hnext + (size_t)c * D_OUT + 4 * lane;
  atomicAdd(dst + 0, w * h.x);
  atomicAdd(dst + 1, w * h.y);
  atomicAdd(dst + 2, w * h.z);
  atomicAdd(dst + 3, w * h.w);
}

// 8) Z = tilde_Z * std_t + mean_t  -> d_out
__global__ __launch_bounds__(256) void restore(
    const float* __restrict__ h, const int* __restrict__ ntype,
    const float* __restrict__ mean, const float* __restrict__ stdv,
    float* __restrict__ Z) {
  const int idx = blockIdx.x * 256 + threadIdx.x;
  if (idx >= N_NODES * D_OUT) return;
  const int i = idx >> 7, d = idx & (D_OUT - 1);
  const int t = ntype[i];
  Z[idx] = h[idx] * stdv[t * D_OUT + d] + mean[t * D_OUT + d];
}

// ---------------------------------------------------------------------------
// 9) contrastive loss terms: sum exp(cos(tilde_a, tilde_b)/TEMP)
// ---------------------------------------------------------------------------
__global__ __launch_bounds__(256) void edge_loss(
    const int* __restrict__ a, const int* __restrict__ b,
    const float* __restrict__ tilde, const float* __restrict__ inv_norm,
    float* __restrict__ acc, int slot) {
  __shared__ float ls[8];
  const int e    = blockIdx.x * 8 + (threadIdx.x >> 5);
  const int lane = threadIdx.x & 31;
  float p = 0.0f;
  if (e < NUM_E) {
    const int ia = a[e], ib = b[e];
    const float4 x = *(const float4*)(tilde + (size_t)ia * D_OUT + 4 * lane);
    const float4 y = *(const float4*)(tilde + (size_t)ib * D_OUT + 4 * lane);
    float d = x.x * y.x + x.y * y.y + x.z * y.z + x.w * y.w;
    d = wave_reduce_sum(d);
    d *= inv_norm[ia] * inv_norm[ib];
    p = expf(d / TEMP);
  }
  if (lane == 0) ls[threadIdx.x >> 5] = p;
  __syncthreads();
  if (threadIdx.x == 0) {
    float s = 0.0f;
#pragma unroll
    for (int i = 0; i < 8; ++i) s += ls[i];
    atomicAdd(&acc[slot], s);
  }
}

__global__ void finalize_loss(const float* __restrict__ acc, float* __restrict__ out) {
  if (threadIdx.x == 0 && blockIdx.x == 0) {
    const float pos = acc[0], neg = acc[1];
    out[0] = -logf(pos / (pos + neg));
  }
}

// ---------------------------------------------------------------------------
extern "C" void kernel_launch(void* const* d_in, const int* in_sizes, int n_in,
                              void* d_out, int out_size, void* d_ws, size_t ws_size,
                              hipStream_t stream) {
  const float* X    = (const float*)d_in[0];   // [N, 256]
  const float* W    = (const float*)d_in[1];   // [128, 256]
  const float* bias = (const float*)d_in[2];   // [128]
  const int*   edge = (const int*)d_in[3];     // [2, E] flat
  const int*   nege = (const int*)d_in[4];     // [2, E] flat
  const int*   ntyp = (const int*)d_in[5];     // [N]
  float* Z = (float*)d_out;                    // N*128 values + 1 loss

  float* p = (float*)d_ws;
  float* tilde = p; p += (size_t)N_NODES * D_OUT;   // H -> l2(H) -> tilde_H
  float* hA    = p; p += (size_t)N_NODES * D_OUT;
  float* hB    = p; p += (size_t)N_NODES * D_OUT;
  float* s1    = p; p += NUM_TYPES * D_OUT;
  float* s2    = p; p += NUM_TYPES * D_OUT;
  float* mean  = p; p += NUM_TYPES * D_OUT;
  float* stdv  = p; p += NUM_TYPES * D_OUT;
  float* dis   = p; p += N_NODES;                   // deg, then rsqrt(deg)
  float* invn  = p; p += N_NODES;
  float* acc   = p; p += 2;

  const int ROWS_BLK  = (N_NODES + 7) / 8;            // wave-per-row grids
  const int ELEM_BLK  = (N_NODES * D_OUT + 255) / 256;
  const int NODE_BLK  = (N_NODES + 255) / 256;
  const int EDGE_BLK  = (NUM_E + 255) / 256;
  const int EWAVE_BLK = (NUM_E + 7) / 8;

  // encoding
  gemm_wmma_f32<<<N_NODES / 16, 256, 0, stream>>>(X, W, bias, tilde);
  l2norm_rows<<<ROWS_BLK, 256, 0, stream>>>(tilde);

  // per-type standardization
  type_stats<<<D_OUT, 256, 0, stream>>>(tilde, ntyp, s1, s2);
  finalize_stats<<<1, NUM_TYPES * D_OUT, 0, stream>>>(s1, s2, mean, stdv);
  standardize<<<ROWS_BLK, 256, 0, stream>>>(tilde, ntyp, mean, stdv, invn);

  // GCN normalization factors
  init_misc<<<NODE_BLK, 256, 0, stream>>>(dis, acc);
  count_deg<<<EDGE_BLK, 256, 0, stream>>>(edge + NUM_E, dis);
  deg_to_dis<<<NODE_BLK, 256, 0, stream>>>(dis);

  // APPNP: K hops, ping-pong hA/hB, hop0 reads tilde
  const float* cur = tilde;
  for (int k = 0; k < K_HOPS; ++k) {
    float* nxt = (k & 1) ? hB : hA;
    appnp_self<<<ELEM_BLK, 256, 0, stream>>>(tilde, cur, dis, nxt);
    appnp_scatter<<<EWAVE_BLK, 256, 0, stream>>>(edge, edge + NUM_E, dis, cur, nxt);
    cur = nxt;
  }

  // restore + loss
  restore<<<ELEM_BLK, 256, 0, stream>>>(cur, ntyp, mean, stdv, Z);
  edge_loss<<<EWAVE_BLK, 256, 0, stream>>>(edge, edge + NUM_E, tilde, invn, acc, 0);
  edge_loss<<<EWAVE_BLK, 256, 0, stream>>>(nege, nege + NUM_E, tilde, invn, acc, 1);
  finalize_loss<<<1, 1, 0, stream>>>(acc, Z + (size_t)N_NODES * D_OUT);
}